// TableTransform_38998303047874
// MI455X (gfx1250) — compile-verified
//
#include <hip/hip_runtime.h>
#include <hip/hip_bf16.h>
#include <math.h>

typedef __attribute__((ext_vector_type(2))) float v2f;
typedef __attribute__((ext_vector_type(8))) float v8f;

namespace {
constexpr int    T_   = 16384;
constexpr int    C_   = 131072;
constexpr int    H_   = 128;
constexpr int    GH_  = 8;      // GAT heads
constexpr int    TCH  = 32;     // TC_HID
constexpr int    AH_  = 64;     // AGGR_HEADS
constexpr int    PH_  = 16;     // P_HEADS
constexpr int    NP_  = 2;
constexpr int    TF_  = 3;
constexpr int    OH_  = 24;
constexpr int    DE_  = 32;
constexpr int    KIN  = TF_ + OH_ + DE_;   // 59
constexpr int    FINK = H_ * NP_ + TF_;    // 259
constexpr float  BIG_ = 65536.0f;

// ---- workspace layout (float element offsets; EMAX/PE hold uint keys) ----
constexpr size_t OFF_TEMB  = 0;                               // [T,H]
constexpr size_t OFF_CTEMB = OFF_TEMB  + (size_t)T_*H_;       // [T,H]
constexpr size_t OFF_S     = OFF_CTEMB + (size_t)T_*H_;       // [T,128]  (h*16+j)
constexpr size_t OFF_CSEMB = OFF_S     + (size_t)T_*H_;       // [C,32]
constexpr size_t OFF_TC    = OFF_CSEMB + (size_t)C_*TCH;      // [T,2048]
constexpr size_t OFF_EL    = OFF_TC    + (size_t)T_*TCH*AH_;  // [C,8]  logits -> e_exp (reused)
constexpr size_t OFF_ER    = OFF_EL    + (size_t)C_*GH_;      // [T,8]
constexpr size_t OFF_DEN   = OFF_ER    + (size_t)T_*GH_;      // [T,8]
constexpr size_t OFF_EMAX  = OFF_DEN   + (size_t)T_*GH_;      // [T,8]  uint keys
constexpr size_t OFF_PE    = OFF_EMAX  + (size_t)T_*GH_;      // [T,32] uint keys
constexpr size_t OFF_UR    = OFF_PE    + (size_t)T_*NP_*PH_;  // [8,128]
constexpr size_t OFF_VL    = OFF_UR    + (size_t)GH_*H_;      // [8,16]
constexpr size_t OFF_WPN   = OFF_VL    + (size_t)GH_*16;      // [128,128] permuted W_gat_src
constexpr size_t OFF_G0    = OFF_WPN   + (size_t)H_*H_;       // [128,16]
constexpr size_t OFF_G1    = OFF_G0    + (size_t)H_*PH_;      // [128,16]
} // namespace

// monotone float <-> uint key (for atomic max on signed floats)
__device__ __forceinline__ unsigned fkey(float f) {
  unsigned u = __float_as_uint(f);
  return (u & 0x80000000u) ? ~u : (u | 0x80000000u);
}
__device__ __forceinline__ float funkey(unsigned k) {
  return __uint_as_float((k & 0x80000000u) ? (k & 0x7FFFFFFFu) : ~k);
}
__device__ __forceinline__ float lrelu(float x, float s) { return x >= 0.f ? x : s * x; }

// fp32 WMMA: D(16x16) = A(16x4) * B(4x16) + C   -> v_wmma_f32_16x16x4_f32
__device__ __forceinline__ v8f wmma4(v2f a, v2f b, v8f c) {
  return __builtin_amdgcn_wmma_f32_16x16x4_f32(false, a, false, b, (short)0, c,
                                               false, false);
}

// ---------------- init ----------------
__global__ void k_init(float* denom, float* S, unsigned* emax, unsigned* pe) {
  size_t i = (size_t)blockIdx.x * blockDim.x + threadIdx.x;
  const unsigned NEGKEY = 0x00800000u;  // fkey(-FLT_MAX)
  if (i < (size_t)T_ * GH_)        { denom[i] = 0.f; emax[i] = NEGKEY; }
  if (i < (size_t)T_ * H_)         S[i] = 0.f;
  if (i < (size_t)T_ * NP_ * PH_)  pe[i] = NEGKEY;
}

// ---------------- small precomputes ----------------
// u_r[h,j] = sum_k attn_r[h,k] * W_gat_dst[h*128+k, j]   (folds h_dst GEMM into er)
// v_l[h,j] = sum_k attn_l[h,k] * W_gat_src[h*128+k, j]   (folds h_src GEMM into el)
__global__ void k_prep_attn(const float* attn_r, const float* Wdst,
                            const float* attn_l, const float* Wsrc,
                            float* u_r, float* v_l) {
  int i = blockIdx.x * blockDim.x + threadIdx.x;
  if (i < GH_ * H_) {
    int h = i / H_, j = i % H_;
    float s = 0.f;
    for (int k = 0; k < H_; ++k) s += attn_r[h * H_ + k] * Wdst[(h * H_ + k) * H_ + j];
    u_r[i] = s;
  } else if (i < GH_ * H_ + GH_ * 16) {
    int r = i - GH_ * H_;
    int h = r / 16, j = r % 16;
    float s = 0.f;
    for (int k = 0; k < H_; ++k) s += attn_l[h * H_ + k] * Wsrc[(h * H_ + k) * 16 + j];
    v_l[r] = s;
  }
}

// Wpn[n, h*16+j] = W_gat_src[h*128+n, j]  (weight for ct_emb = S @ Wpn^T form)
__global__ void k_prep_wp(const float* Wsrc, float* Wpn) {
  int i = blockIdx.x * blockDim.x + threadIdx.x;
  if (i >= H_ * H_) return;
  int n = i >> 7, idx = i & 127;
  int h = idx >> 4, j = idx & 15;
  Wpn[n * H_ + idx] = Wsrc[(h * H_ + n) * 16 + j];
}

// G0[o,p] = sum_i W_final[o,i]     * W_pe0[i,p]
// G1[o,p] = sum_i W_final[o,128+i] * W_pe1[i,p]
__global__ void k_prep_g(const float* Wfin, const float* Wpe0, const float* Wpe1,
                         float* G0, float* G1) {
  int i = blockIdx.x * blockDim.x + threadIdx.x;
  if (i >= 2 * H_ * PH_) return;
  int which = i / (H_ * PH_);
  int r = i % (H_ * PH_);
  int o = r / PH_, p = r % PH_;
  float s = 0.f;
  if (which == 0) {
    for (int k = 0; k < H_; ++k) s += Wfin[o * FINK + k] * Wpe0[k * PH_ + p];
    G0[r] = s;
  } else {
    for (int k = 0; k < H_; ++k) s += Wfin[o * FINK + H_ + k] * Wpe1[k * PH_ + p];
    G1[r] = s;
  }
}

// ---------------- WMMA GEMMs (wave-per-16x16-tile, K-step 4) ----------------
// table_emb = concat(tf,oh,de) @ W_table_all^T      (A gathered from 3 arrays)
__device__ __forceinline__ float tbl_in(const float* tf, const float* oh,
                                        const float* de, int t, int k) {
  if (k < TF_)       return tf[t * TF_ + k];
  if (k < TF_ + OH_) return oh[t * OH_ + (k - TF_)];
  return de[t * DE_ + (k - TF_ - OH_)];
}

__global__ void k_gemm_table_emb(const float* tf, const float* oh, const float* de,
                                 const float* W /*[H,59]*/, float* out /*[T,H]*/) {
  int wave = (blockIdx.x * blockDim.x + threadIdx.x) >> 5;
  int tilesN = H_ >> 4;                         // 8
  int m0 = (wave / tilesN) << 4;
  int n0 = (wave % tilesN) << 4;
  if (m0 >= T_) return;
  int half = (threadIdx.x >> 4) & 1, lr = threadIdx.x & 15;
  v8f acc = {};
  for (int k0 = 0; k0 < KIN; k0 += 4) {
    int ka = k0 + 2 * half;                     // A VGPR0: K=0|2, VGPR1: K=1|3
    v2f a, b;
    a.x = (ka     < KIN) ? tbl_in(tf, oh, de, m0 + lr, ka)     : 0.f;
    a.y = (ka + 1 < KIN) ? tbl_in(tf, oh, de, m0 + lr, ka + 1) : 0.f;
    b.x = (ka     < KIN) ? W[(n0 + lr) * KIN + ka]     : 0.f;
    b.y = (ka + 1 < KIN) ? W[(n0 + lr) * KIN + ka + 1] : 0.f;
    acc = wmma4(a, b, acc);
  }
#pragma unroll
  for (int i = 0; i < 8; ++i)
    out[(size_t)(m0 + i + 8 * half) * H_ + (n0 + lr)] = acc[i];
}

// generic: out[M,N] = scale * (A[M,K] @ W[N,K]^T) + bias[N] + resid[M,N]
__global__ void k_gemm_wmma(const float* A, const float* W, const float* bias,
                            const float* resid, float scale,
                            int M, int N, int K, float* out) {
  int wave = (blockIdx.x * blockDim.x + threadIdx.x) >> 5;
  int tilesN = N >> 4;
  int m0 = (wave / tilesN) << 4;
  int n0 = (wave % tilesN) << 4;
  if (m0 >= M) return;
  int half = (threadIdx.x >> 4) & 1, lr = threadIdx.x & 15;
  const float* arow = A + (size_t)(m0 + lr) * K;
  const float* wrow = W + (size_t)(n0 + lr) * K;
  v8f acc = {};
  for (int k0 = 0; k0 < K; k0 += 4) {
    int ka = k0 + 2 * half;
    v2f a, b;
    a.x = (ka     < K) ? arow[ka]     : 0.f;
    a.y = (ka + 1 < K) ? arow[ka + 1] : 0.f;
    b.x = (ka     < K) ? wrow[ka]     : 0.f;
    b.y = (ka + 1 < K) ? wrow[ka + 1] : 0.f;
    acc = wmma4(a, b, acc);
  }
  float bn = bias ? bias[n0 + lr] : 0.f;
#pragma unroll
  for (int i = 0; i < 8; ++i) {
    int m = m0 + i + 8 * half;
    float v = acc[i] * scale + bn;
    if (resid) v += resid[(size_t)m * N + (n0 + lr)];
    out[(size_t)m * N + (n0 + lr)] = v;
  }
}

// ---------------- attention logits ----------------
__global__ void k_el(const float* cs, const float* v_l, float* el) {
  int c = blockIdx.x * blockDim.x + threadIdx.x;
  if (c >= C_) return;
  float v[16];
#pragma unroll
  for (int j = 0; j < 16; ++j) v[j] = cs[(size_t)c * 16 + j];
#pragma unroll
  for (int h = 0; h < GH_; ++h) {
    float s = 0.f;
#pragma unroll
    for (int j = 0; j < 16; ++j) s += v[j] * v_l[h * 16 + j];
    el[(size_t)c * GH_ + h] = s;
  }
}

__global__ void k_er(const float* temb, const float* u_r, float* er) {
  int i = blockIdx.x * blockDim.x + threadIdx.x;
  if (i >= T_ * GH_) return;
  int t = i >> 3, h = i & 7;
  float s = 0.f;
  for (int j = 0; j < H_; ++j) s += temb[(size_t)t * H_ + j] * u_r[h * H_ + j];
  er[i] = s;
}

// ---------------- edge softmax (3 passes, L2 atomics) ----------------
__global__ void k_edge_max(const int* dst, const float* el, const float* er,
                           unsigned* emax) {
  int c = blockIdx.x * blockDim.x + threadIdx.x;
  if (c >= C_) return;
  int d = dst[c];
#pragma unroll
  for (int h = 0; h < GH_; ++h) {
    float e = lrelu(el[(size_t)c * GH_ + h] + er[d * GH_ + h], 0.2f);
    atomicMax(&emax[d * GH_ + h], fkey(e));
  }
}

__global__ void k_edge_exp(const int* dst, float* el /*in: logits, out: exp*/,
                           const float* er, const unsigned* emax, float* denom) {
  int c = blockIdx.x * blockDim.x + threadIdx.x;
  if (c >= C_) return;
  int d = dst[c];
#pragma unroll
  for (int h = 0; h < GH_; ++h) {
    float e = lrelu(el[(size_t)c * GH_ + h] + er[d * GH_ + h], 0.2f);
    float x = __expf(e - funkey(emax[d * GH_ + h]));
    el[(size_t)c * GH_ + h] = x;
    atomicAdd(&denom[d * GH_ + h], x);
  }
}

// S[t, h*16+j] += alpha[c,h] * cs[c,j]   (linearity folds msg-aggregation)
__global__ void k_scatter_S(const int* dst, const float* eexp, const float* denom,
                            const float* cs, float* S) {
  int i = blockIdx.x * blockDim.x + threadIdx.x;
  if (i >= C_ * GH_) return;
  int c = i >> 3, h = i & 7;
  int d = dst[c];
  float a = eexp[i] / denom[d * GH_ + h];
#pragma unroll
  for (int j = 0; j < 16; ++j)
    atomicAdd(&S[(size_t)d * H_ + h * 16 + j], a * cs[(size_t)c * 16 + j]);
}

// ---------------- per-edge bilinear + predicate heads (wave per edge) ----------
__global__ void k_edge_out(const int* dst, const float* cs_emb, const float* tc,
                           const float* W_ph, const float* b_ph,
                           const float* pf, const int* pm, unsigned* pe) {
  int e = blockIdx.x * (blockDim.x >> 5) + (threadIdx.x >> 5);
  int lane = threadIdx.x & 31;
  if (e >= C_) return;
  int d = dst[e];
  const float* tcrow = tc + (size_t)d * (TCH * AH_);
  __builtin_prefetch(tcrow + lane * 64, 0, 1);     // pull gathered 8KB row
  float csr = cs_emb[(size_t)e * TCH + lane];      // lane j holds cs_emb[e,j]
  float eo0 = 0.f, eo1 = 0.f;                      // lane k owns cols k, k+32
#pragma unroll
  for (int j = 0; j < TCH; ++j) {
    float cj = __shfl(csr, j, 32);
    eo0 += cj * tcrow[j * AH_ + lane];
    eo1 += cj * tcrow[j * AH_ + 32 + lane];
  }
  eo0 = lrelu(eo0, 0.01f);
  eo1 = lrelu(eo1, 0.01f);
  // ph[m] = b_ph[m] + sum_k W_ph[m,k] * act[k]   (m = lane, 32 outputs)
  float acc = b_ph[lane];
  const float* wrow = W_ph + lane * AH_;
#pragma unroll
  for (int k = 0; k < 32; ++k) {
    acc += wrow[k]      * __shfl(eo0, k, 32);
    acc += wrow[32 + k] * __shfl(eo1, k, 32);
  }
  int np = lane >> 4;                              // flat 32 = [NP=2, PH=16]
  float pfv = pf[(size_t)e * NP_ + np];
  float pmv = (float)pm[(size_t)e * NP_ + np];
  float res = pfv * acc - (1.0f - pmv) * BIG_;
  atomicMax(&pe[d * (NP_ * PH_) + lane], fkey(res));
}

// ---------------- final: pe -> (G0,G1 folded) output ----------------
__global__ void k_final(const unsigned* pe, const float* G0, const float* G1,
                        const float* Wfin, const float* tf, float* out) {
  int i = blockIdx.x * blockDim.x + threadIdx.x;
  if (i >= T_ * H_) return;
  int t = i >> 7, o = i & 127;
  float acc = 0.f;
#pragma unroll
  for (int p = 0; p < PH_; ++p) {
    float v0 = funkey(pe[t * (NP_ * PH_) + p]);
    float v1 = funkey(pe[t * (NP_ * PH_) + PH_ + p]);
    v0 = (v0 > -BIG_ * 0.5f) ? v0 : 0.f;
    v1 = (v1 > -BIG_ * 0.5f) ? v1 : 0.f;
    acc += v0 * G0[o * PH_ + p] + v1 * G1[o * PH_ + p];
  }
#pragma unroll
  for (int f = 0; f < TF_; ++f)
    acc += Wfin[o * FINK + H_ * NP_ + f] * tf[t * TF_ + f];
  out[(size_t)t * H_ + o] = acc;
}

extern "C" void kernel_launch(void* const* d_in, const int* in_sizes, int n_in,
                              void* d_out, int out_size, void* d_ws, size_t ws_size,
                              hipStream_t stream) {
  const float* table_features = (const float*)d_in[0];
  const float* table_onehot   = (const float*)d_in[1];
  const float* table_dense    = (const float*)d_in[2];
  const float* colstats       = (const float*)d_in[3];
  const float* pred_feat      = (const float*)d_in[4];
  const int*   pred_mask      = (const int*)d_in[5];
  const int*   edge_dst       = (const int*)d_in[7];   // edge_src == arange(C)
  const float* W_table_all    = (const float*)d_in[8];
  const float* W_gat_src      = (const float*)d_in[9];
  const float* W_gat_dst      = (const float*)d_in[10];
  const float* attn_l         = (const float*)d_in[11];
  const float* attn_r         = (const float*)d_in[12];
  const float* W_cs           = (const float*)d_in[13];
  const float* W_tc           = (const float*)d_in[14];
  const float* b_tc           = (const float*)d_in[15];
  const float* W_ph           = (const float*)d_in[16];
  const float* b_ph           = (const float*)d_in[17];
  const float* W_pe0          = (const float*)d_in[18];
  const float* W_pe1          = (const float*)d_in[19];
  const float* W_final        = (const float*)d_in[20];

  float*    ws    = (float*)d_ws;
  float*    temb  = ws + OFF_TEMB;
  float*    ctemb = ws + OFF_CTEMB;
  float*    S     = ws + OFF_S;
  float*    csemb = ws + OFF_CSEMB;
  float*    tc    = ws + OFF_TC;
  float*    el    = ws + OFF_EL;
  float*    er    = ws + OFF_ER;
  float*    denom = ws + OFF_DEN;
  unsigned* emax  = (unsigned*)(ws + OFF_EMAX);
  unsigned* pe    = (unsigned*)(ws + OFF_PE);
  float*    u_r   = ws + OFF_UR;
  float*    v_l   = ws + OFF_VL;
  float*    Wpn   = ws + OFF_WPN;
  float*    G0    = ws + OFF_G0;
  float*    G1    = ws + OFF_G1;
  float*    out   = (float*)d_out;

  const int B = 256;

  // init + small precomputes
  k_init<<<(T_ * H_ + B - 1) / B, B, 0, stream>>>(denom, S, emax, pe);
  k_prep_attn<<<(GH_ * H_ + GH_ * 16 + B - 1) / B, B, 0, stream>>>(
      attn_r, W_gat_dst, attn_l, W_gat_src, u_r, v_l);
  k_prep_wp<<<(H_ * H_ + B - 1) / B, B, 0, stream>>>(W_gat_src, Wpn);
  k_prep_g<<<(2 * H_ * PH_ + B - 1) / B, B, 0, stream>>>(W_final, W_pe0, W_pe1, G0, G1);

  // table embedding GEMM [T,59]x[59,128]  (WMMA f32)
  k_gemm_table_emb<<<(T_ / 16) * (H_ / 16) * 32 / B, B, 0, stream>>>(
      table_features, table_onehot, table_dense, W_table_all, temb);

  // attention logits
  k_er<<<(T_ * GH_ + B - 1) / B, B, 0, stream>>>(temb, u_r, er);
  k_el<<<(C_ + B - 1) / B, B, 0, stream>>>(colstats, v_l, el);

  // edge softmax + linear message aggregation into S
  k_edge_max<<<(C_ + B - 1) / B, B, 0, stream>>>(edge_dst, el, er, emax);
  k_edge_exp<<<(C_ + B - 1) / B, B, 0, stream>>>(edge_dst, el, er, emax, denom);
  k_scatter_S<<<(C_ * GH_ + B - 1) / B, B, 0, stream>>>(edge_dst, el, denom, colstats, S);

  // ct_emb = table_emb + (1/8) * S @ Wpn^T        [T,128]x[128,128] WMMA
  k_gemm_wmma<<<(T_ / 16) * (H_ / 16) * 32 / B, B, 0, stream>>>(
      S, Wpn, nullptr, temb, 0.125f, T_, H_, H_, ctemb);

  // cs_emb = colstats @ W_cs^T                    [C,16]x[16,32] WMMA
  k_gemm_wmma<<<(C_ / 16) * (TCH / 16) * 32 / B, B, 0, stream>>>(
      colstats, W_cs, nullptr, nullptr, 1.0f, C_, TCH, 16, csemb);

  // tc = ct_emb @ W_tc^T + b_tc                   [T,128]x[128,2048] WMMA (8.6 GF)
  k_gemm_wmma<<<(T_ / 16) * ((TCH * AH_) / 16) * 32 / B, B, 0, stream>>>(
      ctemb, W_tc, b_tc, nullptr, 1.0f, T_, TCH * AH_, H_, tc);

  // per-edge bilinear + predicate heads + segment-max
  k_edge_out<<<C_ / (B / 32), B, 0, stream>>>(edge_dst, csemb, tc, W_ph, b_ph,
                                              pred_feat, pred_mask, pe);

  // final fused projection
  k_final<<<(T_ * H_ + B - 1) / B, B, 0, stream>>>(pe, G0, G1, W_final,
                                                   table_features, out);
}